// Rx_layer_16140487098682
// MI455X (gfx1250) — compile-verified
//
#include <hip/hip_runtime.h>

typedef float v2f __attribute__((ext_vector_type(2)));
typedef float v8f __attribute__((ext_vector_type(8)));

static __device__ __forceinline__ v8f wmma_f32(v2f a, v2f b, v8f c) {
    // D = A(16x4) x B(4x16) + C, f32, wave32
    return __builtin_amdgcn_wmma_f32_16x16x4_f32(false, a, false, b, (short)0, c, false, false);
}

// One workgroup per (batch n, block b) state vector of 2048 reals.
// 11 qubits split 4 + 4 + 3:
//   A = kron(rx0..rx3)  (16x16 complex), contract over bits 10..7  -> WMMA
//   B = kron(rx4..rx7)  (16x16 complex), contract over bits 6..3   -> WMMA
//   qubits 8..10: pairwise butterflies on bits 2..0 in LDS (VALU)
// Closed form: kron entry (i,j) = (-i)^popcount(i^j) * prod(a_q or b_q).
__global__ void __launch_bounds__(256, 1)
rx_layer_kernel(const float* __restrict__ state,
                const float* __restrict__ weights,
                float* __restrict__ out)
{
    const int v      = blockIdx.x;      // n*8 + b
    const int b      = v & 7;
    const int tid    = threadIdx.x;     // 0..255, 8 waves
    const int lane   = tid & 31;
    const int wv     = tid >> 5;        // wave id -> 16-column tile
    const int n16    = lane & 15;       // A row (M) / B col (N)
    const int hi16   = lane >> 4;       // 0/1 half-wave
    const int khalf  = hi16 * 2;        // K offset within 4-wide slice
    const int rowoff = hi16 * 8;        // D row offset

    __shared__ float aqs[11], bqs[11];  // per-qubit cos / sin
    __shared__ float Ar[16][16],  Ai[16][16];
    __shared__ float Brm[16][16], Bim[16][16], nBim[16][16];
    __shared__ float T1re[2048], T1im[2048];   // layout [M'][H*8+L']
    __shared__ float T2re[2048], T2im[2048];   // layout [M ][H*8+L']

    // Per-lane trig (vector libm emitted once, full f32 precision) -> LDS table.
    if (tid < 11) {
        const float h = 0.5f * weights[b * 11 + tid];
        aqs[tid] = cosf(h);
        bqs[tid] = sinf(h);
    }
    __syncthreads();

    // Build A (qubits 0-3) and B (qubits 4-7) complex kron matrices in LDS.
    {
        const int i = tid >> 4, j = tid & 15;
        const int diff = i ^ j;
        float mA = 1.0f, mB = 1.0f;
        #pragma unroll
        for (int t = 0; t < 4; ++t) {           // bit t of index <-> qubit (3-t)+base
            const bool d = (diff >> t) & 1;
            mA *= d ? bqs[3 - t] : aqs[3 - t];
            mB *= d ? bqs[7 - t] : aqs[7 - t];
        }
        const int h = __popc(diff) & 3;         // (-i)^h
        const float cr = (h == 0) ? 1.0f : ((h == 2) ? -1.0f : 0.0f);
        const float ci = (h == 1) ? -1.0f : ((h == 3) ? 1.0f : 0.0f);
        Ar[i][j]  = mA * cr;  Ai[i][j]  = mA * ci;
        Brm[i][j] = mB * cr;  Bim[i][j] = mB * ci;  nBim[i][j] = -(mB * ci);
    }
    __syncthreads();

    const float* S = state + (size_t)v * 2048;   // real input, row H' stride 128
    const int col = wv * 16 + n16;               // this lane's output column

    // ---- Stage 1: T1[H, M'*8+L'] = A @ S   (complex 16x16 times real 16x128)
    {
        v8f accR = {0,0,0,0,0,0,0,0}, accI = {0,0,0,0,0,0,0,0};
        #pragma unroll
        for (int ks = 0; ks < 4; ++ks) {
            const int k0 = ks * 4 + khalf;
            v2f aR, aI, bS;
            aR.x = Ar[n16][k0];  aR.y = Ar[n16][k0 + 1];
            aI.x = Ai[n16][k0];  aI.y = Ai[n16][k0 + 1];
            bS.x = S[k0 * 128 + col];
            bS.y = S[(k0 + 1) * 128 + col];
            accR = wmma_f32(aR, bS, accR);
            accI = wmma_f32(aI, bS, accI);
        }
        // Scatter D tile with permutation: col = M'*8+L' -> [M'][H*8+L']
        const int mp = col >> 3, lp = col & 7;
        #pragma unroll
        for (int r = 0; r < 8; ++r) {
            const int H = r + rowoff;
            const int idx = mp * 128 + H * 8 + lp;
            T1re[idx] = accR[r];
            T1im[idx] = accI[r];
        }
    }
    __syncthreads();

    // ---- Stage 2: T2[M, H*8+L'] = B @ T1   (complex x complex, 4 chains)
    {
        v8f accR = {0,0,0,0,0,0,0,0}, accI = {0,0,0,0,0,0,0,0};
        #pragma unroll
        for (int ks = 0; ks < 4; ++ks) {
            const int k0 = ks * 4 + khalf;
            v2f bR, bI, bNI, tR, tI;
            bR.x  = Brm[n16][k0];  bR.y  = Brm[n16][k0 + 1];
            bI.x  = Bim[n16][k0];  bI.y  = Bim[n16][k0 + 1];
            bNI.x = nBim[n16][k0]; bNI.y = nBim[n16][k0 + 1];
            tR.x = T1re[k0 * 128 + col]; tR.y = T1re[(k0 + 1) * 128 + col];
            tI.x = T1im[k0 * 128 + col]; tI.y = T1im[(k0 + 1) * 128 + col];
            accR = wmma_f32(bR,  tR, accR);   // Re += Br*T1re
            accR = wmma_f32(bNI, tI, accR);   // Re += (-Bi)*T1im
            accI = wmma_f32(bR,  tI, accI);   // Im += Br*T1im
            accI = wmma_f32(bI,  tR, accI);   // Im += Bi*T1re
        }
        #pragma unroll
        for (int r = 0; r < 8; ++r) {
            const int M = r + rowoff;
            const int idx = M * 128 + col;    // [M][H*8+L']
            T2re[idx] = accR[r];
            T2im[idx] = accI[r];
        }
    }

    // ---- Stage 3: qubits 8,9,10 as butterflies on the low 3 LDS-index bits.
    // z0' = a*z0 - i*b*z1 ; z1' = a*z1 - i*b*z0
    #pragma unroll
    for (int g = 0; g < 3; ++g) {
        __syncthreads();
        const int stride = 4 >> g;            // qubit 8 -> bit2, 9 -> bit1, 10 -> bit0
        const float a = aqs[8 + g], bb = bqs[8 + g];
        for (int p = tid; p < 1024; p += 256) {
            const int p0 = ((p & ~(stride - 1)) << 1) | (p & (stride - 1));
            const int p1 = p0 + stride;
            const float r0 = T2re[p0], i0 = T2im[p0];
            const float r1 = T2re[p1], i1 = T2im[p1];
            T2re[p0] = a * r0 + bb * i1;
            T2im[p0] = a * i0 - bb * r1;
            T2re[p1] = a * r1 + bb * i0;
            T2im[p1] = a * i1 - bb * r0;
        }
    }
    __syncthreads();

    // ---- Final: un-permute [M][H*8+L] -> x = H*128 + M*8 + L, interleave re/im.
    float* O = out + (size_t)v * 4096;
    for (int x = tid; x < 2048; x += 256) {
        const int H = x >> 7, M = (x >> 3) & 15, L = x & 7;
        const int idx = M * 128 + H * 8 + L;
        O[2 * x]     = T2re[idx];
        O[2 * x + 1] = T2im[idx];
    }
}

extern "C" void kernel_launch(void* const* d_in, const int* in_sizes, int n_in,
                              void* d_out, int out_size, void* d_ws, size_t ws_size,
                              hipStream_t stream) {
    (void)in_sizes; (void)n_in; (void)d_ws; (void)ws_size; (void)out_size;
    const float* state   = (const float*)d_in[0];   // (32,8,1,2048,1) f32
    const float* weights = (const float*)d_in[1];   // (1,8,1,11,1,1)  f32
    float* out = (float*)d_out;                     // (32,8,1,2048,1) c64 -> interleaved f32
    rx_layer_kernel<<<dim3(256), dim3(256), 0, stream>>>(state, weights, out);
}